// TransformerLayer_73778948211311
// MI455X (gfx1250) — compile-verified
//
#include <hip/hip_runtime.h>
#include <math.h>

#define H    4096
#define NH   32
#define HD   128
#define LMAX 8192
#define EPS  1e-5f

typedef __attribute__((ext_vector_type(2))) float f32x2;
typedef __attribute__((ext_vector_type(8))) float f32x8;

// ---------------------------------------------------------------------------
// D = A(16x4) * B(4x16) + C, fp32 WMMA. We use only row 0 of A/D.
// A layout (32-bit 16x4): lanes 0-15 hold A[M=lane][K=0,1] in V0,V1;
//                         lanes 16-31 hold A[M=lane-16][K=2,3].
// B layout (4x16):        lanes 0-15 hold B[K=0][N],B[K=1][N] in V0,V1;
//                         lanes 16-31 hold B[K=2][N],B[K=3][N].
// D layout: V0 lanes 0-15 = row M=0, N=lane.
// ---------------------------------------------------------------------------
__device__ __forceinline__ f32x8 wmma_f32_4(f32x2 a, f32x2 b, f32x8 c) {
    return __builtin_amdgcn_wmma_f32_16x16x4_f32(false, a, false, b,
                                                 (short)0, c, false, false);
}

__device__ __forceinline__ float ntload(const float* p) {
    return __builtin_nontemporal_load(p);
}
__device__ __forceinline__ f32x2 ntload2(const float* p) {
    return __builtin_nontemporal_load((const f32x2*)p);
}

// ---------------------------------------------------------------------------
// LayerNorm over H elements, one block of 256 threads.
// ---------------------------------------------------------------------------
__global__ __launch_bounds__(256)
void ln_kernel(const float* __restrict__ x, const float* __restrict__ g,
               const float* __restrict__ b, float* __restrict__ y) {
    __shared__ float red[256];
    const int tid = threadIdx.x;

    float s = 0.f;
    for (int i = tid; i < H; i += 256) s += x[i];
    red[tid] = s; __syncthreads();
    for (int off = 128; off > 0; off >>= 1) {
        if (tid < off) red[tid] += red[tid + off];
        __syncthreads();
    }
    const float mean = red[0] * (1.0f / H);
    __syncthreads();

    float vs = 0.f;
    for (int i = tid; i < H; i += 256) { float d = x[i] - mean; vs += d * d; }
    red[tid] = vs; __syncthreads();
    for (int off = 128; off > 0; off >>= 1) {
        if (tid < off) red[tid] += red[tid + off];
        __syncthreads();
    }
    const float inv = rsqrtf(red[0] * (1.0f / H) + EPS);

    for (int i = tid; i < H; i += 256)
        y[i] = (x[i] - mean) * inv * g[i] + b[i];
}

// ---------------------------------------------------------------------------
// GEMV: out[n] = epilogue( sum_k x[k]*W[k*N+n] ), W row-major KxN.
// One block (8 waves) per 16 outputs; split-K across waves; dual-accumulator
// WMMA inner loop; LDS cross-wave reduce. MODE: 0 plain, 1 +res[n], 2 GELU.
// ---------------------------------------------------------------------------
template <int MODE>
__global__ __launch_bounds__(256)
void gemv16_kernel(const float* __restrict__ x, const float* __restrict__ W,
                   const float* __restrict__ res, float* __restrict__ out,
                   int K, int N) {
    const int tid  = threadIdx.x;
    const int lane = tid & 31;
    const int wave = tid >> 5;            // 0..7
    const int half = (lane >> 4) & 1;     // 0: K0/K1, 1: K2/K3
    const int col  = lane & 15;           // N within tile / M within A
    const int n0   = blockIdx.x * 16;
    const float sel = (col == 0) ? 1.0f : 0.0f;   // keep only A row 0

    const int kchunk = K >> 3;            // K divisible by 64 in all uses
    const int kbeg = wave * kchunk;

    // per-lane streaming pointers
    const float* xp = x + kbeg + 2 * half;
    const float* wp = W + (size_t)(kbeg + 2 * half) * N + n0 + col;
    const size_t wstep = (size_t)4 * N;   // advance 4 K-rows

    f32x8 c0 = {}, c1 = {};
    for (int i = 0; i < kchunk; i += 8) {
        f32x2 xv0 = *(const f32x2*)(xp);
        f32x2 xv1 = *(const f32x2*)(xp + 4);
        f32x2 b0, b1;
        b0[0] = ntload(wp);
        b0[1] = ntload(wp + N);
        b1[0] = ntload(wp + wstep);
        b1[1] = ntload(wp + wstep + N);
        xp += 8;
        wp += 2 * wstep;
        f32x2 a0 = xv0 * sel;
        f32x2 a1 = xv1 * sel;
        c0 = wmma_f32_4(a0, b0, c0);
        c1 = wmma_f32_4(a1, b1, c1);
    }

    __shared__ float sh[8 * 16];
    if (lane < 16) sh[wave * 16 + col] = c0[0] + c1[0];   // D row 0
    __syncthreads();

    if (tid < 16) {
        float acc = 0.f;
        #pragma unroll
        for (int w = 0; w < 8; ++w) acc += sh[w * 16 + tid];
        float r;
        if (MODE == 1)      r = res[n0 + tid] + acc;
        else if (MODE == 2) r = 0.5f * acc * (1.0f + erff(acc * 0.70710678118654752f));
        else                r = acc;
        out[n0 + tid] = r;
    }
}

// ---------------------------------------------------------------------------
// Fused attention, one block (8 waves, 256 thr) per head.
// Scores q.K^T/sqrt(HD) via WMMA (16 keys/tile), softmax in LDS, ctx = p.V
// via WMMA (16 dims/wave). KV-cache row `index` is sourced from the fresh
// k/v vectors (functional update; inputs untouched).
// ---------------------------------------------------------------------------
__global__ __launch_bounds__(256)
void attn_kernel(const float* __restrict__ qkv, const float* __restrict__ pk,
                 const float* __restrict__ pv, const int* __restrict__ idxp,
                 float* __restrict__ ctx) {
    __shared__ float s[LMAX];     // scores -> probabilities (32 KB)
    __shared__ float red[256];

    const int h    = blockIdx.x;
    const int tid  = threadIdx.x;
    const int lane = tid & 31;
    const int wave = tid >> 5;
    const int half = (lane >> 4) & 1;
    const int col  = lane & 15;
    const float sel = (col == 0) ? 1.0f : 0.0f;

    const int L  = *idxp + 1;             // valid keys (<= LMAX)
    const int Lp = (L + 15) & ~15;        // padded to tile

    const float* q  = qkv + h * HD;
    const float* kn = qkv + H + h * HD;
    const float* vn = qkv + 2 * H + h * HD;
    const float scale = rsqrtf((float)HD);

    // ---- scores ----
    const int ntiles = Lp >> 4;
    for (int t = wave; t < ntiles; t += 8) {
        const int row = t * 16 + col;               // key index = N
        const int r2  = (row < L) ? row : 0;        // keep reads in-bounds
        const float* krow = (r2 == L - 1) ? kn
                          : pk + ((size_t)h * LMAX + r2) * HD;
        const float* qp = q + 2 * half;
        const float* kp = krow + 2 * half;
        f32x8 c0 = {}, c1 = {};
        #pragma unroll
        for (int kk = 0; kk < HD; kk += 8) {
            f32x2 qv0 = *(const f32x2*)(qp + kk);
            f32x2 qv1 = *(const f32x2*)(qp + kk + 4);
            f32x2 b0  = ntload2(kp + kk);
            f32x2 b1  = ntload2(kp + kk + 4);
            c0 = wmma_f32_4(qv0 * sel, b0, c0);
            c1 = wmma_f32_4(qv1 * sel, b1, c1);
        }
        if (lane < 16) {
            const int rr = t * 16 + lane;
            s[rr] = (rr < L) ? (c0[0] + c1[0]) * scale : -INFINITY;
        }
    }
    __syncthreads();

    // ---- softmax: max ----
    float m = -INFINITY;
    for (int i = tid; i < Lp; i += 256) m = fmaxf(m, s[i]);
    red[tid] = m; __syncthreads();
    for (int off = 128; off > 0; off >>= 1) {
        if (tid < off) red[tid] = fmaxf(red[tid], red[tid + off]);
        __syncthreads();
    }
    m = red[0]; __syncthreads();

    // ---- softmax: exp + sum ----
    float sum = 0.f;
    for (int i = tid; i < Lp; i += 256) {
        float p = __expf(s[i] - m);   // exp(-inf - m) = 0 for padded rows
        s[i] = p;
        sum += p;
    }
    red[tid] = sum; __syncthreads();
    for (int off = 128; off > 0; off >>= 1) {
        if (tid < off) red[tid] += red[tid + off];
        __syncthreads();
    }
    const float inv = 1.0f / red[0];

    // ---- ctx = p . V  (wave w covers head dims [16w, 16w+16)) ----
    const int d0 = wave * 16;
    const float* pvh = pv + (size_t)h * LMAX * HD;
    f32x8 c0 = {}, c1 = {};
    for (int k = 0; k < Lp; k += 8) {
        const int r0 = k + 2 * half;          // even -> 8B-aligned LDS reads
        f32x2 p0 = *(const f32x2*)(s + r0);
        f32x2 p1 = *(const f32x2*)(s + r0 + 4);
        const float* v0 = (r0 == L - 1)     ? vn : pvh + (size_t)r0 * HD;
        const float* v1 = (r0 + 1 == L - 1) ? vn : pvh + (size_t)(r0 + 1) * HD;
        const float* v2 = (r0 + 4 == L - 1) ? vn : pvh + (size_t)(r0 + 4) * HD;
        const float* v3 = (r0 + 5 == L - 1) ? vn : pvh + (size_t)(r0 + 5) * HD;
        f32x2 b0, b1;
        b0[0] = ntload(v0 + d0 + col);
        b0[1] = ntload(v1 + d0 + col);
        b1[0] = ntload(v2 + d0 + col);
        b1[1] = ntload(v3 + d0 + col);
        c0 = wmma_f32_4(p0 * sel, b0, c0);
        c1 = wmma_f32_4(p1 * sel, b1, c1);
    }
    if (lane < 16) ctx[h * HD + d0 + lane] = (c0[0] + c1[0]) * inv;
}

// ---------------------------------------------------------------------------
extern "C" void kernel_launch(void* const* d_in, const int* in_sizes, int n_in,
                              void* d_out, int out_size, void* d_ws, size_t ws_size,
                              hipStream_t stream) {
    const float* x      = (const float*)d_in[0];
    const float* past_k = (const float*)d_in[1];
    const float* past_v = (const float*)d_in[2];
    const float* w_qkv  = (const float*)d_in[3];
    const float* w_out  = (const float*)d_in[4];
    const float* w_ffn1 = (const float*)d_in[5];
    const float* w_ffn2 = (const float*)d_in[6];
    const float* ln1_g  = (const float*)d_in[7];
    const float* ln1_b  = (const float*)d_in[8];
    const float* ln2_g  = (const float*)d_in[9];
    const float* ln2_b  = (const float*)d_in[10];
    const int*   index  = (const int*)d_in[11];

    float* out = (float*)d_out;
    float* ws  = (float*)d_ws;

    float* xn1 = ws;                 // H
    float* qkv = xn1 + H;            // 3H
    float* ctx = qkv + 3 * H;        // H
    float* x2  = ctx + H;            // H
    float* xn2 = x2 + H;             // H
    float* f1  = xn2 + H;            // 4H

    ln_kernel<<<1, 256, 0, stream>>>(x, ln1_g, ln1_b, xn1);

    gemv16_kernel<0><<<(3 * H) / 16, 256, 0, stream>>>(xn1, w_qkv, nullptr, qkv, H, 3 * H);

    attn_kernel<<<NH, 256, 0, stream>>>(qkv, past_k, past_v, index, ctx);

    gemv16_kernel<1><<<H / 16, 256, 0, stream>>>(ctx, w_out, x, x2, H, H);

    ln_kernel<<<1, 256, 0, stream>>>(x2, ln2_g, ln2_b, xn2);

    gemv16_kernel<2><<<(4 * H) / 16, 256, 0, stream>>>(xn2, w_ffn1, nullptr, f1, H, 4 * H);

    gemv16_kernel<1><<<H / 16, 256, 0, stream>>>(f1, w_ffn2, x2, out, 4 * H, H);
}